// FakeNewsGNN_40888088658252
// MI455X (gfx1250) — compile-verified
//
#include <hip/hip_runtime.h>

#define N_NODES 50000
#define N_EDGES 600000
#define D 128
#define OUTD 2
#define N_LAYERS 6
#define SCAN_B 256

typedef __attribute__((ext_vector_type(2))) float v2f;
typedef __attribute__((ext_vector_type(8))) float v8f;

// ---------------------------------------------------------------- utilities
__global__ void zero_i32_kernel(int* __restrict__ p, int n) {
  int i = blockIdx.x * blockDim.x + threadIdx.x;
  int stride = gridDim.x * blockDim.x;
  for (; i < n; i += stride) p[i] = 0;
}

__global__ void count_kernel(const int* __restrict__ dst, int* __restrict__ counts, int e) {
  int i = blockIdx.x * blockDim.x + threadIdx.x;
  if (i < e) atomicAdd(&counts[dst[i]], 1);
}

// ------------------------------------------------ exclusive scan (3 kernels)
__global__ __launch_bounds__(SCAN_B)
void scan1_kernel(const int* __restrict__ counts, int* __restrict__ rowptr,
                  int* __restrict__ partials, int n) {
  __shared__ int sm[SCAN_B];
  const int tid = threadIdx.x;
  const int gid = blockIdx.x * SCAN_B + tid;
  const int v = (gid < n) ? counts[gid] : 0;
  sm[tid] = v;
  __syncthreads();
  for (int off = 1; off < SCAN_B; off <<= 1) {
    int t = (tid >= off) ? sm[tid - off] : 0;
    __syncthreads();
    sm[tid] += t;
    __syncthreads();
  }
  if (gid < n) rowptr[gid] = sm[tid] - v;  // exclusive
  if (tid == SCAN_B - 1) partials[blockIdx.x] = sm[tid];
}

__global__ __launch_bounds__(SCAN_B)
void scan2_kernel(int* __restrict__ partials, int nb) {
  __shared__ int sm[SCAN_B];
  const int tid = threadIdx.x;
  const int v = (tid < nb) ? partials[tid] : 0;
  sm[tid] = v;
  __syncthreads();
  for (int off = 1; off < SCAN_B; off <<= 1) {
    int t = (tid >= off) ? sm[tid - off] : 0;
    __syncthreads();
    sm[tid] += t;
    __syncthreads();
  }
  if (tid < nb) partials[tid] = sm[tid] - v;  // exclusive block offsets
}

__global__ __launch_bounds__(SCAN_B)
void scan3_kernel(int* __restrict__ rowptr, const int* __restrict__ partials,
                  int n, int total) {
  const int gid = blockIdx.x * SCAN_B + threadIdx.x;
  if (gid < n) rowptr[gid] += partials[blockIdx.x];
  if (gid == 0) rowptr[n] = total;
}

__global__ void copy_i32_kernel(const int* __restrict__ a, int* __restrict__ b, int n) {
  int i = blockIdx.x * blockDim.x + threadIdx.x;
  if (i < n) b[i] = a[i];
}

// bucket edges by destination: eid[] holds source node ids grouped by dst
__global__ void fill_kernel(const int* __restrict__ src, const int* __restrict__ dst,
                            int* __restrict__ cursor, int* __restrict__ eid, int e) {
  int i = blockIdx.x * blockDim.x + threadIdx.x;
  if (i < e) {
    int p = atomicAdd(&cursor[dst[i]], 1);
    eid[p] = src[i];
  }
}

// ---------------------- fused SAGE layer: CSR gather-mean + WMMA f32 GEMM
// hout = relu( mean_nbr(hin) @ Wl + hin @ Wr + bl )
// 256 threads = 8 waves, block owns 32 node rows (two 16-row M-tiles).
// Phase 1: each wave gather-means 4 node rows into LDS (h is L2-resident).
// Phase 2: wave w owns output cols [16w,16w+16) for BOTH M-tiles: the two
// accumulators share every loaded B operand -> 4 WMMAs per 4 weight dwords.
__global__ __launch_bounds__(256)
void sage_layer_kernel(const float* __restrict__ hin,
                       const int* __restrict__ rowptr,
                       const int* __restrict__ eid,
                       const float* __restrict__ Wl,
                       const float* __restrict__ Wr,
                       const float* __restrict__ bl,
                       float* __restrict__ hout) {
  __shared__ float xs[32 * D];  // h tile
  __shared__ float gs[32 * D];  // neighbor-mean tile

  const int r0 = blockIdx.x * 32;
  const int t = threadIdx.x;
  const int wave = t >> 5;
  const int lane = t & 31;
  const int c4 = lane << 2;     // float4 chunk: 32 lanes cover 128 cols

  // ---- Phase 1: gather + mean (4 rows per wave); zero-fill tail rows
#pragma unroll
  for (int rv = 0; rv < 4; ++rv) {
    const int rr = wave * 4 + rv;
    const int gr = r0 + rr;
    float4 a = make_float4(0.f, 0.f, 0.f, 0.f);
    float4 xv = make_float4(0.f, 0.f, 0.f, 0.f);
    if (gr < N_NODES) {
      const int beg = rowptr[gr];
      const int end = rowptr[gr + 1];
      for (int j = beg; j < end; ++j) {
        const int s = eid[j];  // wave-uniform
        const float4 v = *(const float4*)(hin + (long)s * D + c4);
        a.x += v.x; a.y += v.y; a.z += v.z; a.w += v.w;
      }
      const float dinv = 1.0f / fmaxf((float)(end - beg), 1.0f);
      a.x *= dinv; a.y *= dinv; a.z *= dinv; a.w *= dinv;
      xv = *(const float4*)(hin + (long)gr * D + c4);
    }
    *(float4*)(gs + rr * D + c4) = a;
    *(float4*)(xs + rr * D + c4) = xv;
  }
  __syncthreads();

  // ---- Phase 2: dual-tile dual-GEMM; B registers reused by both M-tiles
  const int c0 = wave * 16;
  const int mn = lane & 15;            // A: row M / B,D: col N
  const int ksel = (lane >> 4) << 1;   // lanes 0-15: K+{0,1}; 16-31: K+{2,3}

  v8f acc0 = {};
  v8f acc1 = {};
  for (int ks = 0; ks < D; ks += 4) {
    const int k0 = ks + ksel;
    v2f ag0 = *(const v2f*)(gs + mn * D + k0);
    v2f ah0 = *(const v2f*)(xs + mn * D + k0);
    v2f ag1 = *(const v2f*)(gs + (16 + mn) * D + k0);
    v2f ah1 = *(const v2f*)(xs + (16 + mn) * D + k0);
    v2f bL, bR;
    bL.x = Wl[(long)k0 * D + c0 + mn];
    bL.y = Wl[(long)(k0 + 1) * D + c0 + mn];
    bR.x = Wr[(long)k0 * D + c0 + mn];
    bR.y = Wr[(long)(k0 + 1) * D + c0 + mn];
    acc0 = __builtin_amdgcn_wmma_f32_16x16x4_f32(false, ag0, false, bL,
                                                 (short)0, acc0, false, false);
    acc1 = __builtin_amdgcn_wmma_f32_16x16x4_f32(false, ag1, false, bL,
                                                 (short)0, acc1, false, false);
    acc0 = __builtin_amdgcn_wmma_f32_16x16x4_f32(false, ah0, false, bR,
                                                 (short)0, acc0, false, false);
    acc1 = __builtin_amdgcn_wmma_f32_16x16x4_f32(false, ah1, false, bR,
                                                 (short)0, acc1, false, false);
  }

  // ---- Epilogue: bias + ReLU (C/D layout: VGPR v -> row v / v+8)
  const float bias = bl[c0 + mn];
  const int rbase = r0 + ((lane >> 4) << 3);
#pragma unroll
  for (int v = 0; v < 8; ++v) {
    float val = fmaxf(acc0[v] + bias, 0.0f);
    hout[(long)(rbase + v) * D + c0 + mn] = val;  // rows r0..r0+15 always valid
  }
#pragma unroll
  for (int v = 0; v < 8; ++v) {
    const int rr = rbase + 16 + v;
    if (rr < N_NODES) {
      float val = fmaxf(acc1[v] + bias, 0.0f);
      hout[(long)rr * D + c0 + mn] = val;
    }
  }
}

// ----------------------------------------------------- final FC 128 -> 2
__global__ __launch_bounds__(256)
void fc_kernel(const float* __restrict__ h, const float* __restrict__ W,
               const float* __restrict__ b, float* __restrict__ out) {
  int node = blockIdx.x * (blockDim.x >> 5) + (threadIdx.x >> 5);
  int lane = threadIdx.x & 31;
  if (node >= N_NODES) return;
  float s0 = 0.0f, s1 = 0.0f;
  for (int d = lane; d < D; d += 32) {
    float hv = h[(long)node * D + d];
    s0 += hv * W[d * OUTD + 0];
    s1 += hv * W[d * OUTD + 1];
  }
#pragma unroll
  for (int off = 16; off; off >>= 1) {
    s0 += __shfl_down(s0, off, 32);
    s1 += __shfl_down(s1, off, 32);
  }
  if (lane == 0) {
    out[node * OUTD + 0] = s0 + b[0];
    out[node * OUTD + 1] = s1 + b[1];
  }
}

// ---------------------------------------------------------------- dispatcher
extern "C" void kernel_launch(void* const* d_in, const int* in_sizes, int n_in,
                              void* d_out, int out_size, void* d_ws, size_t ws_size,
                              hipStream_t stream) {
  const float* x   = (const float*)d_in[0];
  const int*   ei  = (const int*)d_in[1];
  const float* Wl  = (const float*)d_in[2];
  const float* Wr  = (const float*)d_in[3];
  const float* bl  = (const float*)d_in[4];
  const float* fcW = (const float*)d_in[5];
  const float* fcb = (const float*)d_in[6];
  float* out = (float*)d_out;

  // workspace: hA | hB | counts | rowptr | cursor | eid   (~54 MB)
  float* hA = (float*)d_ws;
  float* hB = hA + (size_t)N_NODES * D;
  int* counts = (int*)(hB + (size_t)N_NODES * D);
  int* rowptr = counts + N_NODES;        // N_NODES + 1 entries
  int* cursor = rowptr + N_NODES + 1;
  int* eid    = cursor + N_NODES;        // N_EDGES entries

  const int* src = ei;
  const int* dst = ei + N_EDGES;

  // ---- build CSR over dst (once; edges are layer-invariant)
  const int nscan_blocks = (N_NODES + SCAN_B - 1) / SCAN_B;  // 196 <= SCAN_B
  zero_i32_kernel<<<(N_NODES + 255) / 256, 256, 0, stream>>>(counts, N_NODES);
  count_kernel<<<(N_EDGES + 255) / 256, 256, 0, stream>>>(dst, counts, N_EDGES);
  scan1_kernel<<<nscan_blocks, SCAN_B, 0, stream>>>(counts, rowptr, cursor /*partials scratch*/, N_NODES);
  scan2_kernel<<<1, SCAN_B, 0, stream>>>(cursor, nscan_blocks);
  scan3_kernel<<<nscan_blocks, SCAN_B, 0, stream>>>(rowptr, cursor, N_NODES, N_EDGES);
  copy_i32_kernel<<<(N_NODES + 255) / 256, 256, 0, stream>>>(rowptr, cursor, N_NODES);
  fill_kernel<<<(N_EDGES + 255) / 256, 256, 0, stream>>>(src, dst, cursor, eid, N_EDGES);

  // ---- 6 fused layers (gather-mean + dual-tile dual-GEMM + bias + ReLU)
  const int layer_blocks = (N_NODES + 31) / 32;  // 1563 (tail block: 16 rows)
  const float* cur = x;
  for (int l = 0; l < N_LAYERS; ++l) {
    float* nxt = (l & 1) ? hB : hA;
    sage_layer_kernel<<<layer_blocks, 256, 0, stream>>>(
        cur, rowptr, eid,
        Wl + (size_t)l * D * D, Wr + (size_t)l * D * D, bl + (size_t)l * D,
        nxt);
    cur = nxt;
  }

  // ---- classifier head
  fc_kernel<<<(N_NODES + 7) / 8, 256, 0, stream>>>(cur, fcW, fcb, out);
}